// S2V_5815385719435
// MI455X (gfx1250) — compile-verified
//
#include <hip/hip_runtime.h>

#define N_NODES 50000
#define N_EDGES 800000
#define DIM 64
#define N_TILES (N_NODES / 16)   // 3125 exact tiles, no remainder

typedef __attribute__((ext_vector_type(2))) float v2f;
typedef __attribute__((ext_vector_type(8))) float v8f;

// Workspace layout (floats):
//   [0,           N_NODES)      degf   (in-degree as float)
//   [N_NODES,   2*N_NODES)      Sp     (sum of max(w,0) per dst node)
//   [2*N_NODES, 3*N_NODES)      Sn     (sum of max(-w,0) per dst node)
//   [3*N_NODES, 3*N_NODES+64)   v3p = W3 @ relu(W4)
//   [3*N_NODES+64, +128)        v3n = W3 @ relu(-W4)

__global__ void s2v_prep(const float* __restrict__ W3,
                         const float* __restrict__ W4,
                         float* __restrict__ ws) {
    int t = blockIdx.x * blockDim.x + threadIdx.x;
    if (t < 3 * N_NODES) ws[t] = 0.0f;
    if (blockIdx.x == 0 && threadIdx.x < DIM) {
        int n = threadIdx.x;
        float p = 0.0f, q = 0.0f;
#pragma unroll 4
        for (int k = 0; k < DIM; ++k) {
            float w3 = W3[n * DIM + k];
            float w4 = W4[k];
            p += w3 * fmaxf(w4, 0.0f);
            q += w3 * fmaxf(-w4, 0.0f);
        }
        ws[3 * N_NODES + n] = p;
        ws[3 * N_NODES + DIM + n] = q;
    }
}

__global__ void s2v_edge(const int* __restrict__ eidx,
                         const float* __restrict__ ew,
                         float* __restrict__ ws) {
    int e = blockIdx.x * blockDim.x + threadIdx.x;
    if (e >= N_EDGES) return;
    int d = eidx[N_EDGES + e];          // edge_index[1][e] (destination / segment id)
    float w = ew[e];
    atomicAdd(&ws[d], 1.0f);                             // degree
    atomicAdd(&ws[N_NODES + d], fmaxf(w, 0.0f));         // positive weight sum
    atomicAdd(&ws[2 * N_NODES + d], fmaxf(-w, 0.0f));    // negative weight magnitude sum
}

// One wave32 per 16-node tile: D(16x64) = mu_tile(16x64) @ W2^T(64x64), then fused epilogue.
__global__ void __launch_bounds__(256)
s2v_node(const float* __restrict__ mu, const float* __restrict__ x,
         const float* __restrict__ W1, const float* __restrict__ W2,
         const float* __restrict__ ws, float* __restrict__ out) {
    const int wave = threadIdx.x >> 5;
    const int lane = threadIdx.x & 31;
    const int tile = blockIdx.x * 8 + wave;
    if (tile >= N_TILES) return;          // wave-uniform: EXEC stays all-1s for WMMA

    const int base = tile * 16;
    const int row  = lane & 15;           // M index for A-frag, N index for B-frag
    const int koff = (lane >> 4) << 1;    // lanes 0-15 -> K{k,k+1}; lanes 16-31 -> K{k+2,k+3}

    // A fragment source: mu[base+row][k+koff .. +1]    (16x4 f32 A layout)
    const float* Ap = mu + (size_t)(base + row) * DIM + koff;
    // B fragment source: B[k][n] = W2[n][k]  (B = W2^T, 4x16 per step, mirrored layout)
    const float* B0 = W2 + (size_t)(0 * 16 + row) * DIM + koff;
    const float* B1 = W2 + (size_t)(1 * 16 + row) * DIM + koff;
    const float* B2 = W2 + (size_t)(2 * 16 + row) * DIM + koff;
    const float* B3 = W2 + (size_t)(3 * 16 + row) * DIM + koff;

    v8f acc0 = {}, acc1 = {}, acc2 = {}, acc3 = {};
#pragma unroll
    for (int kc = 0; kc < 16; ++kc) {
        v2f a  = *(const v2f*)(Ap + kc * 4);
        v2f b0 = *(const v2f*)(B0 + kc * 4);
        v2f b1 = *(const v2f*)(B1 + kc * 4);
        v2f b2 = *(const v2f*)(B2 + kc * 4);
        v2f b3 = *(const v2f*)(B3 + kc * 4);
        acc0 = __builtin_amdgcn_wmma_f32_16x16x4_f32(false, a, false, b0, (short)0, acc0, false, false);
        acc1 = __builtin_amdgcn_wmma_f32_16x16x4_f32(false, a, false, b1, (short)0, acc1, false, false);
        acc2 = __builtin_amdgcn_wmma_f32_16x16x4_f32(false, a, false, b2, (short)0, acc2, false, false);
        acc3 = __builtin_amdgcn_wmma_f32_16x16x4_f32(false, a, false, b3, (short)0, acc3, false, false);
    }

    const float* degf = ws;
    const float* Sp   = ws + N_NODES;
    const float* Sn   = ws + 2 * N_NODES;
    const float* v3p  = ws + 3 * N_NODES;
    const float* v3n  = ws + 3 * N_NODES + DIM;

    // C/D layout: element i of v8f -> M = i + 8*(lane>=16), N = (lane%16) + 16*nb
    const int mhi = (lane >> 4) * 8;
    float degv[8], xv[8], spv[8], snv[8];
#pragma unroll
    for (int i = 0; i < 8; ++i) {
        int node = base + mhi + i;
        degv[i] = degf[node];
        xv[i]   = x[node];
        spv[i]  = Sp[node];
        snv[i]  = Sn[node];
    }
    float w1v[4], pv[4], nv[4];
#pragma unroll
    for (int nb = 0; nb < 4; ++nb) {
        int n = nb * 16 + row;
        w1v[nb] = W1[n];
        pv[nb]  = v3p[n];
        nv[nb]  = v3n[n];
    }

    v8f accs[4] = {acc0, acc1, acc2, acc3};
#pragma unroll
    for (int nb = 0; nb < 4; ++nb) {
#pragma unroll
        for (int i = 0; i < 8; ++i) {
            float val = xv[i] * w1v[nb] + degv[i] * accs[nb][i]
                      + spv[i] * pv[nb] + snv[i] * nv[nb];
            out[(size_t)(base + mhi + i) * DIM + nb * 16 + row] = fmaxf(val, 0.0f);
        }
    }
}

extern "C" void kernel_launch(void* const* d_in, const int* in_sizes, int n_in,
                              void* d_out, int out_size, void* d_ws, size_t ws_size,
                              hipStream_t stream) {
    const float* mu  = (const float*)d_in[0];
    const float* x   = (const float*)d_in[1];
    const int*   eix = (const int*)d_in[2];
    const float* ew  = (const float*)d_in[3];
    const float* W1  = (const float*)d_in[4];
    const float* W2  = (const float*)d_in[5];
    const float* W3  = (const float*)d_in[6];
    const float* W4  = (const float*)d_in[7];
    float* out = (float*)d_out;
    float* ws  = (float*)d_ws;

    (void)in_sizes; (void)n_in; (void)out_size; (void)ws_size;

    const int zelems = 3 * N_NODES;
    s2v_prep<<<(zelems + 255) / 256, 256, 0, stream>>>(W3, W4, ws);
    s2v_edge<<<(N_EDGES + 255) / 256, 256, 0, stream>>>(eix, ew, ws);
    s2v_node<<<(N_TILES + 7) / 8, 256, 0, stream>>>(mu, x, W1, W2, ws, out);
}